// GConv_54065048323075
// MI455X (gfx1250) — compile-verified
//
#include <hip/hip_runtime.h>

#define IN_F  256
#define OUT_F 256

typedef __attribute__((ext_vector_type(2))) float v2f;
typedef __attribute__((ext_vector_type(4))) float v4f;
typedef __attribute__((ext_vector_type(8))) float v8f;

// ---------------------------------------------------------------------------
// Kernel 1: zero the accumulator (which is d_out, used in-place as agg).
// ---------------------------------------------------------------------------
__global__ __launch_bounds__(256) void gconv_zero(float* __restrict__ p, int n4) {
    int i = blockIdx.x * blockDim.x + threadIdx.x;
    if (i < n4) {
        v4f z = {0.f, 0.f, 0.f, 0.f};
        *(v4f*)(p + (size_t)i * 4) = z;
    }
}

// ---------------------------------------------------------------------------
// Kernel 2: edge-parallel scatter. One wave (32 lanes) per edge:
//   agg[row][:] += w * x[col][:]   (256 floats -> 8 per lane, f32 atomics)
// row/col/w are wave-uniform -> compiler scalarizes those loads (s_load).
// x and agg both fit in the 192MB L2, so the random traffic stays on-chip.
// ---------------------------------------------------------------------------
__global__ __launch_bounds__(256) void gconv_scatter(const float* __restrict__ x,
                                                     const int* __restrict__ rowi,
                                                     const int* __restrict__ coli,
                                                     const float* __restrict__ ew,
                                                     float* agg,
                                                     int n_edges) {
    const int lane = threadIdx.x & 31;
    const int wave = threadIdx.x >> 5;
    const int edge = blockIdx.x * (blockDim.x >> 5) + wave;
    if (edge >= n_edges) return;

    // prefetch edge metadata a few KB ahead (emits global_prefetch_b8)
    if (edge + 2048 < n_edges) {
        __builtin_prefetch(&coli[edge + 2048], 0, 0);
        __builtin_prefetch(&rowi[edge + 2048], 0, 0);
    }

    const int   r = rowi[edge];
    const int   c = coli[edge];
    const float w = ew[edge];

    const float* xs = x   + (size_t)c * IN_F + lane * 8;
    float*       ag = agg + (size_t)r * IN_F + lane * 8;

    v4f x0 = *(const v4f*)(xs);
    v4f x1 = *(const v4f*)(xs + 4);
#pragma unroll
    for (int i = 0; i < 4; ++i) atomicAdd(ag + i,     w * x0[i]);
#pragma unroll
    for (int i = 0; i < 4; ++i) atomicAdd(ag + 4 + i, w * x1[i]);
}

// ---------------------------------------------------------------------------
// Kernel 3: in-place dense GEMM  data[m,:] = data[m,:] @ W^T + b
// using V_WMMA_F32_16X16X4_F32 (full f32, matches the f32 reference).
// Block = 256 threads = 8 waves arranged 2 (M) x 4 (N).
// Block tile = 32 rows x 256 cols. A slab staged in LDS (padded stride 260
// -> conflict-free ds_load_b64 fragment reads). In-place is safe: each block
// reads only its own 32 rows, and only through the LDS copy.
//
// Fragment layouts per CDNA5 ISA 7.12.2:
//   A 16x4 f32 : lane holds A[lane&15][2*(lane>>4) + {0,1}]      (v2f)
//   B 4x16 f32 : lane holds B[2*(lane>>4) + {0,1}][lane&15]      (v2f)
//   C/D 16x16  : lane holds D[v + 8*(lane>>4)][lane&15], v=0..7  (v8f)
// ---------------------------------------------------------------------------
#define LDA 260   // 260 % 64 == 4 -> 16 distinct banks across M lanes

__global__ __launch_bounds__(256) void gconv_gemm(float* data,
                                                  const float* __restrict__ Wm,
                                                  const float* __restrict__ bias,
                                                  int n_nodes) {
    __shared__ float As[32][LDA];

    const int lane = threadIdx.x & 31;
    const int wv   = threadIdx.x >> 5;   // 0..7
    const int wr   = wv >> 2;            // 0..1 : M group
    const int wc   = wv & 3;             // 0..3 : N group
    const int ml   = lane & 15;          // M (or N) index within tile
    const int kh   = lane >> 4;          // K half select
    const int block_row = blockIdx.x * 32;

    // ---- stage the 32x256 A slab into LDS (coalesced float4 loads) ----
    for (int i = threadIdx.x; i < 32 * 64; i += 256) {
        int r  = i >> 6;
        int c4 = (i & 63) << 2;
        int gr = block_row + r;
        v4f v  = {0.f, 0.f, 0.f, 0.f};
        if (gr < n_nodes) v = *(const v4f*)(data + (size_t)gr * IN_F + c4);
        *(v4f*)&As[r][c4] = v;
    }
    __syncthreads();

    v8f acc[4];
#pragma unroll
    for (int t = 0; t < 4; ++t) acc[t] = (v8f){0.f,0.f,0.f,0.f,0.f,0.f,0.f,0.f};

    const float* wbase = Wm + (size_t)(wc * 64 + ml) * IN_F + 2 * kh;
    const float* abase = &As[wr * 16 + ml][2 * kh];

    // ---- K loop: 64 steps of K=4, 4 WMMAs (N tiles) per step ----
#pragma unroll 4
    for (int kk = 0; kk < IN_F; kk += 4) {
        v2f a = *(const v2f*)(abase + kk);          // ds_load_b64
#pragma unroll
        for (int t = 0; t < 4; ++t) {
            v2f bf = *(const v2f*)(wbase + (size_t)t * 16 * IN_F + kk);
            acc[t] = __builtin_amdgcn_wmma_f32_16x16x4_f32(
                false, a, false, bf, (short)0, acc[t], false, false);
        }
    }

    // ---- epilogue: add bias, store in place ----
#pragma unroll
    for (int t = 0; t < 4; ++t) {
        int n  = wc * 64 + t * 16 + ml;
        float bn = bias[n];
#pragma unroll
        for (int v = 0; v < 8; ++v) {
            int gr = block_row + wr * 16 + v + 8 * kh;
            if (gr < n_nodes) data[(size_t)gr * OUT_F + n] = acc[t][v] + bn;
        }
    }
}

// ---------------------------------------------------------------------------
extern "C" void kernel_launch(void* const* d_in, const int* in_sizes, int n_in,
                              void* d_out, int out_size, void* d_ws, size_t ws_size,
                              hipStream_t stream) {
    const float* x  = (const float*)d_in[0];
    const int*   ei = (const int*)d_in[1];   // [2, E] flat: row then col
    const float* ew = (const float*)d_in[2];
    const float* Wm = (const float*)d_in[3];
    const float* b  = (const float*)d_in[4];
    float* out = (float*)d_out;

    const int n_nodes = in_sizes[0] / IN_F;
    const int n_edges = in_sizes[2];

    // 1) zero the accumulator (d_out doubles as agg)
    int n4 = (n_nodes * IN_F) / 4;
    gconv_zero<<<(n4 + 255) / 256, 256, 0, stream>>>(out, n4);

    // 2) scatter: one wave per edge, 8 waves per block
    int blocks = (n_edges + 7) / 8;
    gconv_scatter<<<blocks, 256, 0, stream>>>(x, ei, ei + n_edges, ew, out, n_edges);

    // 3) in-place WMMA GEMM + bias
    gconv_gemm<<<(n_nodes + 31) / 32, 256, 0, stream>>>(out, Wm, b, n_nodes);
}